// TopK_30159260353107
// MI455X (gfx1250) — compile-verified
//
#include <hip/hip_runtime.h>
#include <stdint.h>

// Problem constants (from the reference): x is [4096, 24576] fp32, K = 64.
#define COLS            24576
#define TOPK            64u
#define THREADS         512
#define VECS_PER_THREAD (COLS / 4 / THREADS)   // 12 float4 per thread

// ---------------------------------------------------------------------------
// Order-preserving float -> uint key (larger float => larger unsigned key).
// ---------------------------------------------------------------------------
__device__ __forceinline__ unsigned f2key(float f) {
    unsigned u = __float_as_uint(f);
    return (u & 0x80000000u) ? ~u : (u | 0x80000000u);
}

// ---------------------------------------------------------------------------
// CDNA5 async global -> LDS 128-bit copy (ASYNCcnt-tracked).
// Builtin signature (from hipcc diagnostic): arg0 = v4i* in global AS,
// arg1 = v4i* in LDS AS, arg2 = imm offset, arg3 = cpol.
// ---------------------------------------------------------------------------
typedef __attribute__((__vector_size__(16))) int v4i;
typedef __attribute__((address_space(1))) v4i* g_v4i_p;
typedef __attribute__((address_space(3))) v4i* l_v4i_p;

__device__ __forceinline__ void async_copy_b128(void* lds_dst, const void* gsrc) {
#if __has_builtin(__builtin_amdgcn_global_load_async_to_lds_b128)
    __builtin_amdgcn_global_load_async_to_lds_b128(
        (g_v4i_p)(uintptr_t)gsrc,
        (l_v4i_p)(unsigned)(uintptr_t)lds_dst,   // low 32 bits of flat addr = LDS byte offset
        0, 0);
#else
    unsigned lds_off = (unsigned)(uintptr_t)lds_dst;
    asm volatile("global_load_async_to_lds_b128 %0, %1, off"
                 :
                 : "v"(lds_off), "v"(gsrc)
                 : "memory");
#endif
}

__device__ __forceinline__ void wait_async_zero() {
#if __has_builtin(__builtin_amdgcn_s_wait_asynccnt)
    __builtin_amdgcn_s_wait_asynccnt(0);
#else
    asm volatile("s_wait_asynccnt 0" ::: "memory");
#endif
}

// ---------------------------------------------------------------------------
// One workgroup per row. Row (96KB) lives in LDS; 4-pass radix select finds
// the exact 64th-largest key; one vectorized pass writes relu(top-k)/0.
// ---------------------------------------------------------------------------
__global__ __launch_bounds__(THREADS)
void topk_relu_kernel(const float* __restrict__ x, float* __restrict__ out) {
    extern __shared__ unsigned char smem_raw[];
    float*    s_row  = (float*)smem_raw;                              // 24576 f32 = 96KB
    unsigned* s_hist = (unsigned*)(smem_raw + COLS * sizeof(float));  // 256 bins
    unsigned* s_ctl  = s_hist + 256;  // [0]=fixed prefix, [1]=k remaining, [2]=tie counter

    const int    row  = blockIdx.x;
    const int    tid  = threadIdx.x;
    const float* grow = x + (size_t)row * COLS;

    // ---- Stage 1: async-stream the whole row into LDS (ASYNCcnt path) ----
    const float4* gsrc4 = (const float4*)grow;
    float4*       lrow4 = (float4*)s_row;
#pragma unroll
    for (int j = 0; j < VECS_PER_THREAD; ++j) {
        int v = j * THREADS + tid;
        async_copy_b128((void*)(lrow4 + v), (const void*)(gsrc4 + v));
    }
    if (tid == 0) { s_ctl[0] = 0u; s_ctl[1] = TOPK; s_ctl[2] = 0u; }
    wait_async_zero();       // each wave drains its own async loads
    __syncthreads();         // then cross-wave visibility of LDS

    // ---- Stage 2: 4-pass radix select (MSB byte -> LSB byte) ----
#pragma unroll 1
    for (int pass = 0; pass < 4; ++pass) {
        const int      shift = 24 - pass * 8;
        const unsigned mask  = (pass == 0) ? 0u : (0xFFFFFFFFu << (shift + 8));

        if (tid < 256) s_hist[tid] = 0u;
        __syncthreads();
        const unsigned fixedVal = s_ctl[0];

#pragma unroll 1
        for (int j = 0; j < VECS_PER_THREAD; ++j) {
            int    v = j * THREADS + tid;
            float4 d = lrow4[v];
#pragma unroll
            for (int e = 0; e < 4; ++e) {
                unsigned key = f2key((&d.x)[e]);
                if ((key & mask) == fixedVal)
                    atomicAdd(&s_hist[(key >> shift) & 0xFFu], 1u);
            }
        }
        __syncthreads();

        if (tid == 0) {
            unsigned kRem = s_ctl[1];
            unsigned cum  = 0u;
            for (int b = 255; b >= 0; --b) {
                unsigned c = s_hist[b];
                if (cum + c >= kRem) {
                    s_ctl[1] = kRem - cum;                          // ties still needed at/below this byte
                    s_ctl[0] = fixedVal | ((unsigned)b << shift);   // extend fixed prefix
                    break;
                }
                cum += c;
            }
        }
        __syncthreads();
    }

    const unsigned T    = s_ctl[0];   // exact key of the K-th largest element
    const unsigned need = s_ctl[1];   // how many elements equal to T to keep (>=1)

    // ---- Stage 3: write relu(selected) / 0, vectorized float4 stores ----
    float4* orow4 = (float4*)(out + (size_t)row * COLS);
#pragma unroll 1
    for (int j = 0; j < VECS_PER_THREAD; ++j) {
        int    v = j * THREADS + tid;
        float4 d = lrow4[v];
        float4 o;
#pragma unroll
        for (int e = 0; e < 4; ++e) {
            float    f   = (&d.x)[e];
            unsigned key = f2key(f);
            float    r   = f > 0.0f ? f : 0.0f;
            float    val = 0.0f;
            if (key > T) {
                val = r;
            } else if (key == T) {
                unsigned pos = atomicAdd(&s_ctl[2], 1u);
                if (pos < need) val = r;   // exact-count tie handling
            }
            (&o.x)[e] = val;
        }
        orow4[v] = o;
    }
}

// ---------------------------------------------------------------------------
// Harness entry point.
// ---------------------------------------------------------------------------
extern "C" void kernel_launch(void* const* d_in, const int* in_sizes, int n_in,
                              void* d_out, int out_size, void* d_ws, size_t ws_size,
                              hipStream_t stream) {
    (void)n_in; (void)d_ws; (void)ws_size; (void)out_size;
    const float* x   = (const float*)d_in[0];
    float*       out = (float*)d_out;

    const int rows = in_sizes[0] / COLS;  // 4096
    const size_t shmem = (size_t)COLS * sizeof(float)   // row buffer (96KB)
                       + 256 * sizeof(unsigned)         // histogram
                       + 16 * sizeof(unsigned);         // control words

    topk_relu_kernel<<<dim3(rows), dim3(THREADS), shmem, stream>>>(x, out);
}